// Encoder_34127810134591
// MI455X (gfx1250) — compile-verified
//
#include <hip/hip_runtime.h>
#include <hip/hip_bf16.h>

typedef float v2f __attribute__((ext_vector_type(2)));
typedef float v8f __attribute__((ext_vector_type(8)));

#define DCH 128
#define APAD 132  // padded LDS row stride (dwords): 16B-aligned, banks 4r -> conflict-free

// ---------------------------------------------------------------- zero
__global__ __launch_bounds__(256) void zero_f32(float* __restrict__ p, size_t n) {
    size_t i = (size_t)blockIdx.x * 256 + threadIdx.x;
    if (i < n) p[i] = 0.0f;
}

// ---------------------------------------------------------------- scatter: agg[dst] += x[src]; cnt[dst]++
// one wave per edge; lane handles 4 channels (float4), 4 global fp32 atomics.
// agg_i+agg_d (~102MB) stay L2-resident (192MB), so atomics resolve on-chip.
__global__ __launch_bounds__(256) void sage_scatter(
    const float* __restrict__ xsrc, const int* __restrict__ src,
    const int* __restrict__ dst, float* __restrict__ agg,
    float* __restrict__ cnt, int E) {
    int e = blockIdx.x * 8 + (threadIdx.x >> 5);
    if (e >= E) return;
    int lane = threadIdx.x & 31;
    int s = src[e];
    int d = dst[e];
    const float4 v = *(const float4*)(xsrc + (size_t)s * DCH + lane * 4);
    float* o = agg + (size_t)d * DCH + lane * 4;
    atomicAdd(o + 0, v.x);
    atomicAdd(o + 1, v.y);
    atomicAdd(o + 2, v.z);
    atomicAdd(o + 3, v.w);
    if (lane == 0) atomicAdd(cnt + d, 1.0f);
}

// ---------------------------------------------------------------- fused mean + dual-GEMM + bias + PReLU
// block = 8 waves, covers rows [16*bid,16*bid+16) x all 128 cols.
// A panels (agg rows, x rows) staged to LDS once per block; wave w computes the
// 16x16 tile at cols [16w,16w+16) with v_wmma_f32_16x16x4_f32, keeping separate
// accumulators for the neighbor and root products so the 1/cnt scale is applied
// once in the epilogue instead of per k-step.
__global__ __launch_bounds__(256) void sage_update(
    const float* __restrict__ agg, const float* __restrict__ cnt,
    const float* __restrict__ xdst,
    const float* __restrict__ Wl, const float* __restrict__ bl,
    const float* __restrict__ Wr, const float* __restrict__ alpha,
    float* __restrict__ out, int n_nodes) {
    __shared__ float sAgg[16 * APAD];
    __shared__ float sX[16 * APAD];
    __shared__ float sInv[16];

    const int t = threadIdx.x;
    const int wave = t >> 5;
    const int lane = t & 31;
    const int m0 = blockIdx.x * 16;

    // ---- stage A panels: 16 rows x 128 cols x 2 matrices, float4 per thread x2
#pragma unroll
    for (int q = 0; q < 2; ++q) {
        const int f4 = t + q * 256;    // 0..511
        const int r = f4 >> 5;         // 0..15
        const int c4 = f4 & 31;        // 0..31
        int row = m0 + r;
        if (row >= n_nodes) row = n_nodes - 1;
        const float4 va = *(const float4*)(agg + (size_t)row * DCH + c4 * 4);
        const float4 vx = *(const float4*)(xdst + (size_t)row * DCH + c4 * 4);
        *(float4*)(sAgg + r * APAD + c4 * 4) = va;
        *(float4*)(sX + r * APAD + c4 * 4) = vx;
    }
    if (t < 16) {
        int row = m0 + t;
        if (row >= n_nodes) row = n_nodes - 1;
        sInv[t] = 1.0f / fmaxf(cnt[row], 1.0f);
    }
    __syncthreads();

    const int n0 = wave * 16;
    const int half = lane >> 4;  // K sub-pair select: {k,k+1} vs {k+2,k+3}
    const int mr = lane & 15;    // A row in tile; B/C/D column

    const float* sa = sAgg + mr * APAD;
    const float* sx = sX + mr * APAD;

    v8f accL = {0.f, 0.f, 0.f, 0.f, 0.f, 0.f, 0.f, 0.f};
    v8f accR = {0.f, 0.f, 0.f, 0.f, 0.f, 0.f, 0.f, 0.f};

    for (int k = 0; k < DCH; k += 4) {
        const int ka = k + 2 * half;
        const v2f a1 = *(const v2f*)(sa + ka);  // agg panel (unscaled)
        const v2f a2 = *(const v2f*)(sx + ka);  // root panel
        v2f b1;
        b1.x = Wl[(size_t)ka * DCH + n0 + mr];
        b1.y = Wl[(size_t)(ka + 1) * DCH + n0 + mr];
        v2f b2;
        b2.x = Wr[(size_t)ka * DCH + n0 + mr];
        b2.y = Wr[(size_t)(ka + 1) * DCH + n0 + mr];

        accL = __builtin_amdgcn_wmma_f32_16x16x4_f32(false, a1, false, b1,
                                                     (short)0, accL, false, false);
        accR = __builtin_amdgcn_wmma_f32_16x16x4_f32(false, a2, false, b2,
                                                     (short)0, accR, false, false);
    }

    const int n = n0 + mr;
    const float bias = bl[n];
    const float a = alpha[n];

    if (m0 + 16 <= n_nodes) {  // full tile: straight-line stores
#pragma unroll
        for (int i = 0; i < 8; ++i) {
            const int rowOut = m0 + i + 8 * half;
            float v = fmaf(accL[i], sInv[i + 8 * half], accR[i] + bias);
            v = (v >= 0.0f) ? v : a * v;
            out[(size_t)rowOut * DCH + n] = v;
        }
    } else {  // tail tile
#pragma unroll
        for (int i = 0; i < 8; ++i) {
            const int rowOut = m0 + i + 8 * half;
            float v = fmaf(accL[i], sInv[i + 8 * half], accR[i] + bias);
            v = (v >= 0.0f) ? v : a * v;
            if (rowOut < n_nodes) out[(size_t)rowOut * DCH + n] = v;
        }
    }
}

// ---------------------------------------------------------------- launch
extern "C" void kernel_launch(void* const* d_in, const int* in_sizes, int n_in,
                              void* d_out, int out_size, void* d_ws, size_t ws_size,
                              hipStream_t stream) {
    const float* x_domain = (const float*)d_in[0];
    const float* x_ip     = (const float*)d_in[1];
    const int*   ei_d2i   = (const int*)d_in[2];
    const int*   ei_i2d   = (const int*)d_in[3];
    const float* Wl       = (const float*)d_in[4];
    const float* bl       = (const float*)d_in[5];
    const float* Wr       = (const float*)d_in[6];
    const float* alpha    = (const float*)d_in[7];

    const int D = DCH;
    const int n_dom = in_sizes[0] / D;
    const int n_ip  = in_sizes[1] / D;
    const int Ed2i  = in_sizes[2] / 2;
    const int Ei2d  = in_sizes[3] / 2;

    float* ws = (float*)d_ws;
    const size_t nd = (size_t)n_dom * D;
    const size_t ni = (size_t)n_ip * D;
    // layout: [agg_d | cnt_d | agg_i | cnt_i | xi_buf | xd_buf]
    float* agg_d  = ws;
    float* cnt_d  = agg_d + nd;
    float* agg_i  = cnt_d + n_dom;
    float* cnt_i  = agg_i + ni;
    float* xi_buf = cnt_i + n_ip;
    float* xd_buf = xi_buf + ni;
    float* out    = (float*)d_out;

    const size_t DD = (size_t)D * D;

    // ================= layer 0 =================
    {
        const size_t z = nd + n_dom + ni + n_ip;  // contiguous agg_d,cnt_d,agg_i,cnt_i
        zero_f32<<<(unsigned)((z + 255) / 256), 256, 0, stream>>>(ws, z);
    }
    // domain -> ip aggregation
    sage_scatter<<<(Ed2i + 7) / 8, 256, 0, stream>>>(x_domain, ei_d2i, ei_d2i + Ed2i,
                                                     agg_i, cnt_i, Ed2i);
    // ip -> domain aggregation
    sage_scatter<<<(Ei2d + 7) / 8, 256, 0, stream>>>(x_ip, ei_i2d, ei_i2d + Ei2d,
                                                     agg_d, cnt_d, Ei2d);
    // ip update  (l=0, etype=0)
    sage_update<<<(n_ip + 15) / 16, 256, 0, stream>>>(agg_i, cnt_i, x_ip,
                                                      Wl + 0 * DD, bl + 0 * D,
                                                      Wr + 0 * DD, alpha + 0 * D,
                                                      xi_buf, n_ip);
    // domain update  (l=0, etype=1)
    sage_update<<<(n_dom + 15) / 16, 256, 0, stream>>>(agg_d, cnt_d, x_domain,
                                                       Wl + 1 * DD, bl + 1 * D,
                                                       Wr + 1 * DD, alpha + 0 * D,
                                                       xd_buf, n_dom);

    // ================= layer 1 =================
    // final output is x_domain only -> layer-1 ip update (and its d2i
    // aggregation) is dead code; compute only the domain path.
    {
        const size_t z = nd + n_dom;  // contiguous agg_d, cnt_d
        zero_f32<<<(unsigned)((z + 255) / 256), 256, 0, stream>>>(ws, z);
    }
    sage_scatter<<<(Ei2d + 7) / 8, 256, 0, stream>>>(xi_buf, ei_i2d, ei_i2d + Ei2d,
                                                     agg_d, cnt_d, Ei2d);
    // domain update (l=1, etype=1) -> d_out
    sage_update<<<(n_dom + 15) / 16, 256, 0, stream>>>(agg_d, cnt_d, xd_buf,
                                                       Wl + 3 * DD, bl + 3 * D,
                                                       Wr + 3 * DD, alpha + 1 * D,
                                                       out, n_dom);
}